// IGCN_59966333387103
// MI455X (gfx1250) — compile-verified
//
#include <hip/hip_runtime.h>

typedef float v2f __attribute__((ext_vector_type(2)));
typedef float v8f __attribute__((ext_vector_type(8)));

#define N_NODES   100000
#define F_IN      512
#define HID       64
#define NCLS      64
#define N_EDGES   3200000
#define WAVES     8
#define ROW_TILES ((N_NODES + 15) / 16)   // 6250, exact

// ---------------------------------------------------------------------------
// Kernel 1: zero the output (d_out is poisoned to 0xAA by the harness and we
// accumulate into it with atomics).
// ---------------------------------------------------------------------------
__global__ __launch_bounds__(256) void igcn_zero(float4* __restrict__ out, int n4) {
    int i = blockIdx.x * blockDim.x + threadIdx.x;
    if (i < n4) out[i] = make_float4(0.f, 0.f, 0.f, 0.f);
}

// ---------------------------------------------------------------------------
// Kernel 2: fused MLP  logits = relu(X@W1 + b1)@W2 + b2  using fp32 WMMA.
//
// One wave (32 lanes) owns a 16-row tile of X. Accumulators: 4x v8f = the
// 16x64 output strip (four 16x16 C/D tiles).
//
// A-fragment layout for V_WMMA_F32_16X16X4_F32 (ISA 7.12.2, 32-bit A 16x4):
//   lanes 0-15  : M = lane,    VGPR0 = K=k0+0, VGPR1 = K=k0+1
//   lanes 16-31 : M = lane-16, VGPR0 = K=k0+2, VGPR1 = K=k0+3
// => per lane: float2 load at X[row0 + (lane&15)][k0 + 2*(lane>>4)].
//
// B-fragment (4x16, row striped across lanes within a VGPR):
//   lanes 0-15 : VGPR0 = B[k0+0][N], VGPR1 = B[k0+1][N]
//   lanes16-31 : VGPR0 = B[k0+2][N], VGPR1 = B[k0+3][N]
// => per lane: W[(k0+2*half)*ld + n*16 + l16] and same + ld.
//
// C/D layout: VGPR r -> row (r + 8*half), col = n*16 + l16.
//
// GEMM1 C-fragments are relayed through LDS (per-wave 16x64 strip) so GEMM2
// can re-read them in A-fragment order; LDS ops within a wave are in-order.
// ---------------------------------------------------------------------------
__global__ __launch_bounds__(256) void igcn_mlp(
    const float* __restrict__ X,   // [N_NODES, F_IN]
    const float* __restrict__ W1,  // [F_IN, HID]
    const float* __restrict__ B1,  // [HID]
    const float* __restrict__ W2,  // [HID, NCLS]
    const float* __restrict__ B2,  // [NCLS]
    float* __restrict__ logits)    // [N_NODES, NCLS] (workspace)
{
    __shared__ float hbuf[WAVES * 16 * HID];   // 32 KB

    const int wave = threadIdx.x >> 5;
    const int lane = threadIdx.x & 31;
    const int tile = blockIdx.x * WAVES + wave;
    if (tile >= ROW_TILES) return;             // wave-uniform exit

    const int half = lane >> 4;                // 0 or 1
    const int l16  = lane & 15;
    const int row0 = tile * 16;

    // ---------------- GEMM1: acc = X_tile @ W1 ----------------
    const float* arow = X + (size_t)(row0 + l16) * F_IN + 2 * half;

    v8f acc[4] = {v8f{}, v8f{}, v8f{}, v8f{}};
    for (int k = 0; k < F_IN; k += 4) {
        v2f a = *(const v2f*)(arow + k);                          // global_load_b64
        const float* wk = W1 + (size_t)(k + 2 * half) * HID + l16; // L2-hot
#pragma unroll
        for (int n = 0; n < 4; ++n) {
            v2f b;
            b.x = wk[n * 16];
            b.y = wk[n * 16 + HID];
            acc[n] = __builtin_amdgcn_wmma_f32_16x16x4_f32(
                false, a, false, b, (short)0, acc[n], false, false);
        }
    }

    // ---------------- bias + ReLU -> LDS ----------------
    float* hb = hbuf + wave * (16 * HID);
#pragma unroll
    for (int n = 0; n < 4; ++n) {
        const float bias = B1[n * 16 + l16];
#pragma unroll
        for (int r = 0; r < 8; ++r) {
            float h = acc[n][r] + bias;
            h = fmaxf(h, 0.f);
            hb[(r + 8 * half) * HID + n * 16 + l16] = h;          // ds_store_b32
        }
    }

    // ---------------- GEMM2: acc = h_tile @ W2 ----------------
    // LDS within a wave is processed in order: reads below see the stores above.
    const float* ha = hb + (size_t)l16 * HID + 2 * half;

    v8f acc2[4] = {v8f{}, v8f{}, v8f{}, v8f{}};
    for (int k = 0; k < HID; k += 4) {
        v2f a = *(const v2f*)(ha + k);                            // ds_load_b64
        const float* wk = W2 + (size_t)(k + 2 * half) * NCLS + l16;
#pragma unroll
        for (int n = 0; n < 4; ++n) {
            v2f b;
            b.x = wk[n * 16];
            b.y = wk[n * 16 + NCLS];
            acc2[n] = __builtin_amdgcn_wmma_f32_16x16x4_f32(
                false, a, false, b, (short)0, acc2[n], false, false);
        }
    }

    // ---------------- bias + store logits ----------------
#pragma unroll
    for (int n = 0; n < 4; ++n) {
        const float bias = B2[n * 16 + l16];
#pragma unroll
        for (int r = 0; r < 8; ++r) {
            const int row = row0 + r + 8 * half;
            logits[(size_t)row * NCLS + n * 16 + l16] = acc2[n][r] + bias;
        }
    }
}

// ---------------------------------------------------------------------------
// Kernel 3: COO scatter  out[row] += val * logits[col].
// One wave per edge; each lane owns 2 contiguous channels (float2 gather +
// two hardware f32 atomic adds). logits and out are both L2-resident (25.6 MB
// each vs 192 MB L2), so the random traffic stays on-chip.
// ---------------------------------------------------------------------------
__global__ __launch_bounds__(256) void igcn_edges(
    const int*   __restrict__ rows,
    const int*   __restrict__ cols,
    const float* __restrict__ vals,
    const float* __restrict__ logits,
    float*       __restrict__ out)
{
    const long long gid = (long long)blockIdx.x * blockDim.x + threadIdx.x;
    const int e = (int)(gid >> 5);
    if (e >= N_EDGES) return;
    const int lane = (int)(gid & 31);

    const int   r = rows[e];
    const int   c = cols[e];
    const float v = vals[e];

    const float2 lv = *(const float2*)(logits + (size_t)c * NCLS + 2 * lane);
    float* dst = out + (size_t)r * NCLS + 2 * lane;
#if defined(__HIP_DEVICE_COMPILE__)
    unsafeAtomicAdd(dst + 0, v * lv.x);   // global_atomic_add_f32 (no return)
    unsafeAtomicAdd(dst + 1, v * lv.y);
#else
    atomicAdd(dst + 0, v * lv.x);
    atomicAdd(dst + 1, v * lv.y);
#endif
}

// ---------------------------------------------------------------------------
// Launch
// ---------------------------------------------------------------------------
extern "C" void kernel_launch(void* const* d_in, const int* in_sizes, int n_in,
                              void* d_out, int out_size, void* d_ws, size_t ws_size,
                              hipStream_t stream) {
    const float* X   = (const float*)d_in[0];   // [100000, 512]
    const int*   er  = (const int*)  d_in[1];   // [3.2M]
    const int*   ec  = (const int*)  d_in[2];   // [3.2M]
    const float* ev  = (const float*)d_in[3];   // [3.2M]
    const float* W1  = (const float*)d_in[4];   // [512, 64]
    const float* B1  = (const float*)d_in[5];   // [64]
    const float* W2  = (const float*)d_in[6];   // [64, 64]
    const float* B2  = (const float*)d_in[7];   // [64]
    float* out    = (float*)d_out;              // [100000, 64]
    float* logits = (float*)d_ws;               // needs 25.6 MB scratch

    // 1) zero output accumulator
    const int n4 = (N_NODES * NCLS) / 4;        // 1,600,000 float4
    igcn_zero<<<(n4 + 255) / 256, 256, 0, stream>>>((float4*)out, n4);

    // 2) fused MLP -> logits in workspace
    const int mlp_blocks = (ROW_TILES + WAVES - 1) / WAVES;   // 782
    igcn_mlp<<<mlp_blocks, 256, 0, stream>>>(X, W1, B1, W2, B2, logits);

    // 3) edge scatter-add
    const long long threads = (long long)N_EDGES * 32;
    const int edge_blocks = (int)((threads + 255) / 256);     // 400,000
    igcn_edges<<<edge_blocks, 256, 0, stream>>>(er, ec, ev, logits, out);
}